// Stage2Assigner_36807869726943
// MI455X (gfx1250) — compile-verified
//
#include <hip/hip_runtime.h>

#define BB 32
#define MM 512
#define NN 4096
#define KT 4
#define HN 2048                 // proposals per LDS half-tile
#define HN_ELEMS (HN * 4)       // 8192 f32 elements per half-tile
#define TOT (BB * MM * KT)      // 65536 elements per output tensor

typedef unsigned int uint_t;
typedef __attribute__((ext_vector_type(4))) unsigned int u32x4;
typedef __attribute__((ext_vector_type(8))) unsigned int u32x8;

// ---- shared arithmetic helpers: contraction OFF so both passes produce ----
// ---- bit-identical IoU values (needed for the lq equality test).       ----
__device__ __forceinline__ float4 cvt_box(float4 b) {
#pragma clang fp contract(off)
    float4 r;
    r.x = b.x - 0.5f * b.z;
    r.y = b.y - 0.5f * b.w;
    r.z = b.x + 0.5f * b.z;
    r.w = b.y + 0.5f * b.w;
    return r;
}

__device__ __forceinline__ float iou_xyxy(float4 a, float4 b) {
#pragma clang fp contract(off)
    float area1 = (a.z - a.x) * (a.w - a.y);
    float area2 = (b.z - b.x) * (b.w - b.y);
    float ltx = fmaxf(a.x, b.x);
    float lty = fmaxf(a.y, b.y);
    float rbx = fminf(a.z, b.z);
    float rby = fminf(a.w, b.w);
    float w = fmaxf(rbx - ltx, 0.0f);
    float h = fmaxf(rby - lty, 0.0f);
    float inter = w * h;
    float uni = area1 + area2 - inter;
    return inter / fmaxf(uni, 1e-9f);
}

__device__ __forceinline__ bool better(float v, int i, float v2, int i2) {
    return (v > v2) || (v == v2 && i < i2);   // value desc, index asc (JAX top_k tie rule)
}

// =====================  Pass 1: per-GT top-K + max_per_gt  =====================
// grid = B * (M/8) blocks, 256 threads (8 waves, one wave per gt)
__global__ __launch_bounds__(256) void s2a_pass1(
    const float* __restrict__ gt_boxes, const float* __restrict__ proposals,
    float* __restrict__ topv, int* __restrict__ topi,
    float* __restrict__ mpg, int* __restrict__ counts)
{
    // LDS: [0, 32768) proposals half-tile (2048 boxes * 16B)
    //      [32768, 36864) merge values (8 waves * 128 f32)
    //      [36864, 40960) merge indices
    __shared__ __align__(16) unsigned char smem[HN * 16 + 1024 * 4 + 1024 * 4];
    float* sprop = (float*)smem;
    float* svals = (float*)(smem + HN * 16);
    int*   sinds = (int*)(smem + HN * 16 + 4096);

    const int tid  = threadIdx.x;
    const int b    = blockIdx.x >> 6;
    const int mg   = blockIdx.x & 63;
    const int w    = tid >> 5;
    const int lane = tid & 31;
    const int m    = mg * 8 + w;
    const int bm   = b * MM + m;

    float4 g = cvt_box(((const float4*)gt_boxes)[bm]);

    float lv[KT];
    int   li[KT];
#pragma unroll
    for (int k = 0; k < KT; ++k) { lv[k] = -1.0f; li[k] = 0x7FFFFFFF; }

    const unsigned long long pbase =
        (unsigned long long)(uintptr_t)proposals + (unsigned long long)b * (NN * 16ull);

    for (int half = 0; half < 2; ++half) {
        // ---- TDM: DMA one 32KB proposal half-tile into LDS (wave 0 issues) ----
        if (tid < 32) {
            unsigned long long ga = pbase + (unsigned long long)half * (HN * 16ull);
            // D# group 0: count=1 | lds_addr=0 | global_addr | type=2
            u32x4 g0 = { 1u, 0u, (uint_t)ga,
                         (uint_t)((ga >> 32) & 0x01FFFFFFull) | 0x80000000u };
            // D# group 1: data_size=4B; tensor_dim0=tile_dim0=stride=8192; dim1=tile1=1
            u32x8 g1 = { 0x00020000u,
                         ((uint_t)HN_ELEMS) << 16,   // tensor_dim0[15:0]
                         0x00010000u,                // tensor_dim1 = 1
                         ((uint_t)HN_ELEMS) << 16,   // tile_dim0
                         1u,                         // tile_dim1 = 1
                         (uint_t)HN_ELEMS,           // tensor_dim0_stride
                         0u, 0u };
            asm volatile("tensor_load_to_lds %0, %1" :: "s"(g0), "s"(g1) : "memory");
            __builtin_amdgcn_s_wait_tensorcnt(0);
        }
        __syncthreads();

        // ---- convert staged cxcywh -> xyxy in place ----
        for (int i = tid; i < HN; i += 256) {
            float4 bx = ((float4*)sprop)[i];
            ((float4*)sprop)[i] = cvt_box(bx);
        }
        __syncthreads();

        // ---- per-lane top-4 over this half (n ascending within a lane) ----
        for (int it = 0; it < HN / 32; ++it) {
            int nl = it * 32 + lane;
            float4 p = ((float4*)sprop)[nl];
            float val = iou_xyxy(g, p);
            int n = half * HN + nl;
            if (val > lv[KT - 1]) {
                lv[KT - 1] = val; li[KT - 1] = n;
#pragma unroll
                for (int k = KT - 1; k > 0; --k) {
                    if (lv[k] > lv[k - 1]) {   // strict: equal keeps earlier index first
                        float tv = lv[k]; lv[k] = lv[k - 1]; lv[k - 1] = tv;
                        int   ti = li[k]; li[k] = li[k - 1]; li[k - 1] = ti;
                    }
                }
            }
        }
        __syncthreads();   // all reads done before next half overwrites LDS
    }

    // ---- cross-lane merge (32 lanes * 4 candidates -> final top-4) ----
#pragma unroll
    for (int k = 0; k < KT; ++k) {
        svals[w * 128 + lane * 4 + k] = lv[k];
        sinds[w * 128 + lane * 4 + k] = li[k];
    }
    __syncthreads();

    if (lane == 0) {
        float fv[KT];
        int   fi[KT];
#pragma unroll
        for (int k = 0; k < KT; ++k) { fv[k] = -1.0f; fi[k] = 0x7FFFFFFF; }
        for (int j = 0; j < 128; ++j) {
            float val = svals[w * 128 + j];
            int   idx = sinds[w * 128 + j];
            if (better(val, idx, fv[KT - 1], fi[KT - 1])) {
                fv[KT - 1] = val; fi[KT - 1] = idx;
#pragma unroll
                for (int k = KT - 1; k > 0; --k) {
                    if (better(fv[k], fi[k], fv[k - 1], fi[k - 1])) {
                        float tv = fv[k]; fv[k] = fv[k - 1]; fv[k - 1] = tv;
                        int   ti = fi[k]; fi[k] = fi[k - 1]; fi[k - 1] = ti;
                    }
                }
            }
        }
#pragma unroll
        for (int k = 0; k < KT; ++k) {
            topv[bm * KT + k] = fv[k];
            topi[bm * KT + k] = fi[k];
        }
        mpg[bm]    = fv[0];   // max over N (exact bits of the winning IoU)
        counts[bm] = 0;       // zero for pass 2's atomics (every call)
    }
}

// =====================  Pass 2: per-proposal match + counts  =====================
// grid = B * (N/256) blocks, 256 threads (one proposal per thread)
__global__ __launch_bounds__(256) void s2a_pass2(
    const float* __restrict__ gt_boxes, const float* __restrict__ proposals,
    const float* __restrict__ mpg, int* __restrict__ counts)
{
    // LDS: [0, 8192) gt boxes (512 * 16B), [8192, 10240) max_per_gt (512 f32)
    __shared__ __align__(16) unsigned char smem[MM * 16 + MM * 4];
    float* sgt  = (float*)smem;
    float* smax = (float*)(smem + MM * 16);

    const int tid = threadIdx.x;
    const int b   = blockIdx.x >> 4;
    const int n   = ((blockIdx.x & 15) << 8) | tid;

    // ---- async global -> LDS staging of gt boxes + max_per_gt ----
    {
        unsigned long long gg =
            (unsigned long long)(uintptr_t)gt_boxes + (unsigned long long)b * (MM * 16ull);
        for (int i = 0; i < 2; ++i) {
            uint_t off = (uint_t)(i * 256 + tid) * 16u;
            unsigned long long ga = gg + off;
            asm volatile("global_load_async_to_lds_b128 %0, %1, off"
                         :: "v"(off), "v"(ga) : "memory");
        }
        uint_t loff = (uint_t)(MM * 16) + (uint_t)tid * 8u;
        unsigned long long ga = (unsigned long long)(uintptr_t)mpg +
                                (unsigned long long)b * (MM * 4ull) + (uint_t)tid * 8u;
        asm volatile("global_load_async_to_lds_b64 %0, %1, off"
                     :: "v"(loff), "v"(ga) : "memory");
        asm volatile("s_wait_asynccnt 0x0" ::: "memory");
    }
    __syncthreads();

    for (int i = tid; i < MM; i += 256) {
        float4 bx = ((float4*)sgt)[i];
        ((float4*)sgt)[i] = cvt_box(bx);
    }
    __syncthreads();

    float4 p = cvt_box(((const float4*)proposals)[(size_t)b * NN + n]);

    float best = -1.0f;
    int   bidx = 0;
    bool  lq   = false;
    for (int m = 0; m < MM; ++m) {
        float4 gb = ((float4*)sgt)[m];       // LDS broadcast (uniform addr)
        float v = iou_xyxy(gb, p);
        if (v > best) { best = v; bidx = m; }  // strict > keeps first max (jnp.argmax)
        lq = lq || (v == smax[m]);             // bit-exact match with pass 1
    }
    if (lq || best >= 0.6f) {
        atomicAdd(&counts[b * MM + bidx], 1);
    }
}

// =====================  Pass 3: emit outputs  =====================
// grid = B*M/256 blocks; out = [scores | pr_inds | gt_inds | valid], all f32
__global__ __launch_bounds__(256) void s2a_pass3(
    const float* __restrict__ topv, const int* __restrict__ topi,
    const int* __restrict__ counts, float* __restrict__ out)
{
    const int bm = blockIdx.x * 256 + threadIdx.x;   // 0 .. B*M-1
    const int c  = counts[bm];
    const int m  = bm & (MM - 1);
#pragma unroll
    for (int k = 0; k < KT; ++k) {
        const bool  val = (k < c);
        const float s   = topv[bm * KT + k];
        const int   pi  = topi[bm * KT + k];
        out[            bm * KT + k] = val ? s : 0.0f;  // scores
        out[    TOT  +  bm * KT + k] = (float)pi;       // pr_inds
        out[2 * TOT  +  bm * KT + k] = (float)m;        // gt_inds
        out[3 * TOT  +  bm * KT + k] = val ? 1.0f : 0.0f; // valid
    }
}

extern "C" void kernel_launch(void* const* d_in, const int* in_sizes, int n_in,
                              void* d_out, int out_size, void* d_ws, size_t ws_size,
                              hipStream_t stream) {
    (void)in_sizes; (void)n_in; (void)out_size; (void)ws_size;

    const float* gt_boxes  = (const float*)d_in[0];   // [B,M,4] f32 cxcywh
    const float* proposals = (const float*)d_in[1];   // [B,N,4] f32 cxcywh
    // d_in[2] = gt_labels (unused by the reference's outputs)
    float* out = (float*)d_out;

    // workspace layout (all fully rewritten every call)
    char*  ws     = (char*)d_ws;
    float* topv   = (float*)(ws);                 // B*M*K f32  (256 KB)
    int*   topi   = (int*)  (ws + 262144);        // B*M*K i32  (256 KB)
    float* mpg    = (float*)(ws + 524288);        // B*M   f32  ( 64 KB)
    int*   counts = (int*)  (ws + 589824);        // B*M   i32  ( 64 KB)

    s2a_pass1<<<BB * (MM / 8), 256, 0, stream>>>(gt_boxes, proposals,
                                                 topv, topi, mpg, counts);
    s2a_pass2<<<BB * (NN / 256), 256, 0, stream>>>(gt_boxes, proposals, mpg, counts);
    s2a_pass3<<<(BB * MM) / 256, 256, 0, stream>>>(topv, topi, counts, out);
}